// Embedder_70360154243390
// MI455X (gfx1250) — compile-verified
//
#include <hip/hip_runtime.h>
#include <hip/hip_bf16.h>
#include <stdint.h>

// Problem constants (fixed by the reference).
#define BB   8
#define TT   8
#define BT   64
#define NN   1024
#define CC   32
#define WSZ  128
#define HW   (WSZ * WSZ)   // 16384

typedef __attribute__((ext_vector_type(2))) float v2f;
typedef __attribute__((ext_vector_type(8))) float v8f;

// ---------------------------------------------------------------------------
// Kernel 1: zero the fp32 accumulators in workspace.
// ---------------------------------------------------------------------------
__global__ void k_zero(float* __restrict__ p, int n) {
    int i = blockIdx.x * blockDim.x + threadIdx.x;
    if (i < n) p[i] = 0.0f;
}

// ---------------------------------------------------------------------------
// Kernel 2: per-frame normalize + segmented mean + atomic scatter.
// One block per frame (bt). One wave32 per segment, lane == channel (C==32).
// Accumulator layout is CELL-MAJOR: acc_w[b][cell][c] so each segment's 32
// atomics land in one contiguous 128B cacheline.
// ---------------------------------------------------------------------------
__global__ void k_scatter(const float* __restrict__ w_part,
                          const int*   __restrict__ sorted_indices,
                          const int*   __restrict__ seq_unique_list,
                          const int*   __restrict__ seq_unique_counts,
                          float* __restrict__ acc_w,   // [B][HW][C]
                          float* __restrict__ acc_m) { // [B][HW]
    const int bt   = blockIdx.x;       // 0..63
    const int b    = bt >> 3;          // bt / T
    const int t    = threadIdx.x;      // 0..255
    const int base = bt * NN;

    __shared__ int s_cnt[NN];
    __shared__ int s_pre[NN];
    __shared__ int s_sum[256];

    // Load counts (coalesced) and per-thread partial sums.
    int c4[4];
#pragma unroll
    for (int i = 0; i < 4; ++i) {
        c4[i] = seq_unique_counts[base + t * 4 + i];
        s_cnt[t * 4 + i] = c4[i];
    }
    s_sum[t] = c4[0] + c4[1] + c4[2] + c4[3];
    __syncthreads();

    // Hillis-Steele inclusive scan over 256 thread totals.
    for (int off = 1; off < 256; off <<= 1) {
        int v = (t >= off) ? s_sum[t - off] : 0;
        __syncthreads();
        s_sum[t] += v;
        __syncthreads();
    }
    int run = (t > 0) ? s_sum[t - 1] : 0;   // exclusive prefix for this thread
#pragma unroll
    for (int i = 0; i < 4; ++i) { s_pre[t * 4 + i] = run; run += c4[i]; }
    __syncthreads();

    const int wave = t >> 5;
    const int lane = t & 31;               // lane == channel

    for (int seg = wave; seg < NN; seg += 8) {
        int cnt = s_cnt[seg];
        if (cnt == 0) break;               // zero counts are a strict suffix
        int start = s_pre[seg];
        float sum = 0.0f;
        for (int p = 0; p < cnt; ++p) {
            int src = sorted_indices[base + start + p];
            const float* row = w_part + (size_t)(base + src) * CC;
            if (p + 1 < cnt) {             // gfx1250 global_prefetch_b8
                int src2 = sorted_indices[base + start + p + 1];
                __builtin_prefetch(w_part + (size_t)(base + src2) * CC, 0, 0);
            }
            float v  = row[lane];          // coalesced 128B per point
            float n2 = v * v;
#pragma unroll
            for (int m = 16; m >= 1; m >>= 1) n2 += __shfl_xor(n2, m, 32);
            float scale = 1.0f / fmaxf(sqrtf(n2), 1e-12f);
            sum = fmaf(v, scale, sum);
        }
        // mean over segment, pre-divided by T for the temporal mean.
        float mean = (sum / (float)cnt) * (1.0f / (float)TT);
        int cell = seq_unique_list[base + seg];
        unsafeAtomicAdd(&acc_w[((size_t)b * HW + cell) * CC + lane], mean);
        if (lane == 0) unsafeAtomicAdd(&acc_m[b * HW + cell], (float)cnt);
    }
}

// ---------------------------------------------------------------------------
// Kernel 3: masked merge with past map + cell-major -> channel-major
// transpose done exactly via a chain of 4x V_WMMA_F32_16X16X4_F32.
//
//   D[ch, cell] = sum_j  A_j(16ch x 4cell) @ B_j(4 x 16, 0/1 selector)
//
// A-layout (ISA 7.12.2, 32-bit A 16x4): lane m (0..15) holds row m with
// K=0,1 in the two A VGPRs; lanes 16..31 hold the same rows with K=2,3.
// B assumed symmetric (VGPR0: K=0 | K=2, VGPR1: K=1 | K=3 across lane halves).
// Products are x1.0/x0.0 with fp32 accumulate -> bit-exact transpose.
// One wave per 16-cell tile; 2 transposes cover the 32 channels.
// ---------------------------------------------------------------------------
__global__ void k_merge(const float* __restrict__ acc_w,   // [B][HW][C]
                        const float* __restrict__ acc_m,   // [B][HW]
                        const float* __restrict__ past_w,  // [B][C][HW]
                        const float* __restrict__ past_m,  // [B][HW]
                        float* __restrict__ out_w,         // [B][C][HW]
                        float* __restrict__ out_m) {       // [B][HW]
    const int wave  = threadIdx.x >> 5;
    const int lane  = threadIdx.x & 31;
    const int tile  = blockIdx.x * 8 + wave;     // 0 .. B*HW/16-1 = 8191
    const int b     = tile >> 10;                // tile / (HW/16)
    const int cellb = (tile & 1023) << 4;        // tile%1024 * 16
    const int n     = lane & 15;                 // post-transpose: lane -> cell
    const int khalf = lane >> 4;                 // A/B K half for this lane
    const int cell  = cellb + n;

    float m  = acc_m[b * HW + cell];
    float pm = past_m[b * HW + cell];
    float denom = m + pm;
    denom = (denom == 0.0f) ? 1.0f : denom;
    if (lane < 16) out_m[b * HW + cell] = denom;
    const float s1 = m / denom;
    const float s2 = pm / denom;

    for (int h = 0; h < 2; ++h) {
        const int chbase = h * 16;
        v8f acc = {};
#pragma unroll
        for (int j = 0; j < 4; ++j) {
            const int cellA = cellb + 4 * j + 2 * khalf;
            const float* p0 = acc_w + ((size_t)b * HW + cellA) * CC + chbase + n;
            v2f A;  A.x = p0[0];  A.y = p0[CC];         // cells cellA, cellA+1
            v2f Bm; Bm.x = (n == 4 * j + 2 * khalf)     ? 1.0f : 0.0f;
                    Bm.y = (n == 4 * j + 2 * khalf + 1) ? 1.0f : 0.0f;
            acc = __builtin_amdgcn_wmma_f32_16x16x4_f32(
                false, A, false, Bm, (short)0, acc, false, false);
        }
        // D layout: VGPR v, lanes 0-15 -> M=v, lanes 16-31 -> M=v+8; N=lane&15.
#pragma unroll
        for (int v = 0; v < 8; ++v) {
            const int ch = chbase + v + 8 * khalf;
            const size_t o = ((size_t)b * CC + ch) * HW + cell;
            out_w[o] = fmaf(acc[v], s1, past_w[o] * s2);   // coalesced along cells
        }
    }
}

// ---------------------------------------------------------------------------
extern "C" void kernel_launch(void* const* d_in, const int* in_sizes, int n_in,
                              void* d_out, int out_size, void* d_ws, size_t ws_size,
                              hipStream_t stream) {
    const float* w_part         = (const float*)d_in[0];
    const int*   sorted_indices = (const int*)d_in[1];
    const int*   seq_unique_list   = (const int*)d_in[2];
    const int*   seq_unique_counts = (const int*)d_in[3];
    const float* past_w         = (const float*)d_in[4];
    const float* past_m         = (const float*)d_in[5];
    // d_in[6] = T (== 8, folded into constants)

    float* out_w = (float*)d_out;                   // [B][C][HW]
    float* out_m = out_w + (size_t)BB * CC * HW;    // [B][1][HW]

    float* acc_w = (float*)d_ws;                    // [B][HW][C]  (16 MB)
    float* acc_m = acc_w + (size_t)BB * HW * CC;    // [B][HW]     (512 KB)

    const int nacc = BB * HW * CC + BB * HW;        // 4,325,376 floats
    k_zero<<<dim3((nacc + 255) / 256), dim3(256), 0, stream>>>(acc_w, nacc);

    k_scatter<<<dim3(BT), dim3(256), 0, stream>>>(
        w_part, sorted_indices, seq_unique_list, seq_unique_counts, acc_w, acc_m);

    // B*HW/16 tiles, 8 waves (tiles) per 256-thread block -> 1024 blocks.
    k_merge<<<dim3(BB * HW / 16 / 8), dim3(256), 0, stream>>>(
        acc_w, acc_m, past_w, past_m, out_w, out_m);
}